// SelfLocalATT_38946763440337
// MI455X (gfx1250) — compile-verified
//
#include <hip/hip_runtime.h>
#include <hip/hip_bf16.h>

typedef __attribute__((ext_vector_type(2))) float v2f;
typedef __attribute__((ext_vector_type(8))) float v8f;

#define EPSF 1e-5f
#define SB_STRIDE 68   // floats; 272B row stride: 16B-aligned, LDS bank stride 4

// D = A*B + C, A:16x4 f32, B:4x16 f32, C/D:16x16 f32 (wave32)
__device__ __forceinline__ v8f wmma4(v2f a, v2f b, v8f c) {
    return __builtin_amdgcn_wmma_f32_16x16x4_f32(
        /*neg_a=*/false, a, /*neg_b=*/false, b,
        /*c_mod=*/(short)0, c, /*reuse_a=*/false, /*reuse_b=*/false);
}

// Async global->LDS copy, 16B per lane (ASYNCcnt-tracked, no VGPR data)
__device__ __forceinline__ void async_ld16(unsigned lds_off, const float* g) {
    asm volatile("global_load_async_to_lds_b128 %0, %1, off"
                 :: "v"(lds_off), "v"(g) : "memory");
}
__device__ __forceinline__ void wait_async_le32() {
    asm volatile("s_wait_asynccnt 0x20" ::: "memory");
}
__device__ __forceinline__ void wait_async_0() {
    asm volatile("s_wait_asynccnt 0x0" ::: "memory");
}

// ---------------------------------------------------------------------------
// 1x1 conv as batched GEMM:  O[nb][m][l] = sum_k W[m][k] * X[nb][k][l]
// W: [128 x 256] row-major, X: [NB x 256 x 1024], O: [NB x 128 x 1024]
// One wave per 16(M) x 64(L) tile.  B staged to LDS via double-buffered
// async b128 loads; fragments read from LDS ([k][col] layout).
// ---------------------------------------------------------------------------
__global__ void k_conv1x1(const float* __restrict__ W, const float* __restrict__ X,
                          float* __restrict__ O) {
    __shared__ __align__(16) float sb[2][64 * SB_STRIDE];
    const int lane = threadIdx.x & 31;
    const int n0 = blockIdx.x * 64;
    const int m0 = blockIdx.y * 16;
    const int nb = blockIdx.z;
    const int ml = lane & 15;
    const int kh = (lane >> 4) * 2;     // 0 for lanes 0-15, 2 for lanes 16-31

    const float* B  = X + (size_t)nb * 256 * 1024;
    float* Cc       = O + (size_t)nb * 128 * 1024;
    const float* arow = W + (size_t)(m0 + ml) * 256 + kh;

    auto stage = [&](int buf, int kbeg) {   // K-slab [kbeg,kbeg+64) x cols [n0,n0+64)
#pragma unroll
        for (int rr = 0; rr < 2; ++rr) {
            const int r = lane + 32 * rr;                       // k_rel
            const float* gp = B + (size_t)(kbeg + r) * 1024 + n0;
            const unsigned lo = (unsigned)(uintptr_t)(&sb[buf][r * SB_STRIDE]);
#pragma unroll
            for (int c = 0; c < 16; ++c) async_ld16(lo + c * 16u, gp + c * 4);
        }
    };

    stage(0, 0);
    v8f acc0{}, acc1{}, acc2{}, acc3{};
    for (int si = 0; si < 4; ++si) {
        const int kbeg = si * 64;
        if (si < 3) { stage((si + 1) & 1, kbeg + 64); wait_async_le32(); }
        else        { wait_async_0(); }
        const float* sbuf = &sb[si & 1][0];
        for (int k0 = 0; k0 < 64; k0 += 4) {
            v2f a; a.x = arow[kbeg + k0]; a.y = arow[kbeg + k0 + 1];
            const float* r0 = sbuf + (k0 + kh) * SB_STRIDE + ml;   // k row
            const float* r1 = r0 + SB_STRIDE;                      // k+1 row
            v2f b;
            b.x = r0[0];  b.y = r1[0];  acc0 = wmma4(a, b, acc0);
            b.x = r0[16]; b.y = r1[16]; acc1 = wmma4(a, b, acc1);
            b.x = r0[32]; b.y = r1[32]; acc2 = wmma4(a, b, acc2);
            b.x = r0[48]; b.y = r1[48]; acc3 = wmma4(a, b, acc3);
        }
    }
    const int mrow = m0 + (lane >> 4) * 8;
    const int col  = n0 + ml;
#pragma unroll
    for (int v = 0; v < 8; ++v) {
        float* cp = Cc + (size_t)(mrow + v) * 1024 + col;
        cp[0]  = acc0[v];
        cp[16] = acc1[v];
        cp[32] = acc2[v];
        cp[48] = acc3[v];
    }
}

// ---------------------------------------------------------------------------
// Local 3x3 attention scores with the f+f (=2f) fold:
//   s[n][r*1024 + l] = 2 * sum_c q[n][c][l] * k[n][c][neigh(l,r)]   (zero pad)
// ---------------------------------------------------------------------------
__global__ void k_scores(const float* __restrict__ q, const float* __restrict__ k,
                         float* __restrict__ s) {
    const int n = blockIdx.y;
    const int l = blockIdx.x * blockDim.x + threadIdx.x;   // 0..1023
    const int i = l >> 5, j = l & 31;
    const float* qn = q + (size_t)n * 128 * 1024;
    const float* kn = k + (size_t)n * 128 * 1024;
    float acc[9] = {};
    for (int c = 0; c < 128; ++c) {
        float qv = qn[c * 1024 + l];
        const float* kc = kn + c * 1024;
#pragma unroll
        for (int r = 0; r < 9; ++r) {
            int di = r / 3 - 1, dj = r % 3 - 1;
            int ii = i + di, jj = j + dj;
            float kv = (ii >= 0 && ii < 32 && jj >= 0 && jj < 32) ? kc[ii * 32 + jj] : 0.f;
            acc[r] = fmaf(qv, kv, acc[r]);
        }
    }
    float* sn = s + (size_t)n * 9216;
#pragma unroll
    for (int r = 0; r < 9; ++r) sn[r * 1024 + l] = 2.0f * acc[r];
}

// ---------------------------------------------------------------------------
// Per-row LayerNorm statistics: mu[n], rstd[n] over D elements
// ---------------------------------------------------------------------------
__global__ void k_lnstats(const float* __restrict__ X, int D,
                          float* __restrict__ mu, float* __restrict__ rstd) {
    const int n = blockIdx.x;
    const float* xn = X + (size_t)n * D;
    float s = 0.f, ss = 0.f;
    for (int i = threadIdx.x; i < D; i += blockDim.x) {
        float v = xn[i];
        s += v; ss = fmaf(v, v, ss);
    }
    __shared__ float sh[256], sh2[256];
    sh[threadIdx.x] = s; sh2[threadIdx.x] = ss;
    __syncthreads();
    for (int w = blockDim.x >> 1; w > 0; w >>= 1) {
        if (threadIdx.x < (unsigned)w) {
            sh[threadIdx.x]  += sh[threadIdx.x + w];
            sh2[threadIdx.x] += sh2[threadIdx.x + w];
        }
        __syncthreads();
    }
    if (threadIdx.x == 0) {
        float m   = sh[0] / (float)D;
        float var = sh2[0] / (float)D - m * m;
        mu[n]   = m;
        rstd[n] = rsqrtf(var + EPSF);
    }
}

// ---------------------------------------------------------------------------
// Elementwise LayerNorm apply: Y[n][d] = (X[n][d]-mu[n])*rstd[n]*g[d] + b[d]
// ---------------------------------------------------------------------------
__global__ void k_ln_affine(const float* __restrict__ X, const float* __restrict__ g,
                            const float* __restrict__ bt, const float* __restrict__ mu,
                            const float* __restrict__ rstd, float* __restrict__ Y, int D) {
    const int n = blockIdx.y;
    const int i = blockIdx.x * blockDim.x + threadIdx.x;
    if (i < D) {
        const float m = mu[n], rs = rstd[n];
        const size_t idx = (size_t)n * D + i;
        Y[idx] = fmaf((X[idx] - m) * rs, g[i], bt[i]);
    }
}

__global__ void k_zero(float* __restrict__ p, int n) {
    int i = blockIdx.x * blockDim.x + threadIdx.x;
    if (i < n) p[i] = 0.f;
}

// ---------------------------------------------------------------------------
// FFN1 (split-K): h[m][j] += sum_{k in chunk} fln[m][k] * w1[j][k]
// M=32 samples, K=9216 (36 chunks of 256 = 4 async slabs each), Nn=256.
// w1 staged via double-buffered async-LDS ([j_rel][k_rel], ds b64 frags).
// ---------------------------------------------------------------------------
__global__ void k_ffn1(const float* __restrict__ fln, const float* __restrict__ w1,
                       float* __restrict__ h) {
    __shared__ __align__(16) float sb[2][64 * SB_STRIDE];
    const int lane = threadIdx.x & 31;
    const int n0 = blockIdx.x * 64;      // output-feature tile (of 256)
    const int m0 = blockIdx.y * 16;      // sample tile (of 32)
    const int kchunk = blockIdx.z * 256; // 36 chunks
    const int ml = lane & 15;
    const int kh = (lane >> 4) * 2;
    const float* arow = fln + (size_t)(m0 + ml) * 9216 + kchunk + kh;

    auto stage = [&](int buf, int kbeg) {   // kbeg relative to kchunk
#pragma unroll
        for (int rr = 0; rr < 2; ++rr) {
            const int r = lane + 32 * rr;                       // j_rel
            const float* gp = w1 + (size_t)(n0 + r) * 9216 + kchunk + kbeg;
            const unsigned lo = (unsigned)(uintptr_t)(&sb[buf][r * SB_STRIDE]);
#pragma unroll
            for (int c = 0; c < 16; ++c) async_ld16(lo + c * 16u, gp + c * 4);
        }
    };

    stage(0, 0);
    v8f acc0{}, acc1{}, acc2{}, acc3{};
    for (int si = 0; si < 4; ++si) {
        const int kbeg = si * 64;
        if (si < 3) { stage((si + 1) & 1, kbeg + 64); wait_async_le32(); }
        else        { wait_async_0(); }
        const float* sbuf = &sb[si & 1][0];
        for (int k0 = 0; k0 < 64; k0 += 4) {
            v2f a; a.x = arow[kbeg + k0]; a.y = arow[kbeg + k0 + 1];
            const int ko = k0 + kh;
            v2f b;
            b = *(const v2f*)(sbuf + (ml)      * SB_STRIDE + ko); acc0 = wmma4(a, b, acc0);
            b = *(const v2f*)(sbuf + (16 + ml) * SB_STRIDE + ko); acc1 = wmma4(a, b, acc1);
            b = *(const v2f*)(sbuf + (32 + ml) * SB_STRIDE + ko); acc2 = wmma4(a, b, acc2);
            b = *(const v2f*)(sbuf + (48 + ml) * SB_STRIDE + ko); acc3 = wmma4(a, b, acc3);
        }
    }
    const int mrow = m0 + (lane >> 4) * 8;
#pragma unroll
    for (int v = 0; v < 8; ++v) {
        float* hp = h + (size_t)(mrow + v) * 256 + n0 + ml;
        atomicAdd(hp + 0,  acc0[v]);
        atomicAdd(hp + 16, acc1[v]);
        atomicAdd(hp + 32, acc2[v]);
        atomicAdd(hp + 48, acc3[v]);
    }
}

__global__ void k_bias_relu(float* __restrict__ h, const float* __restrict__ b1) {
    int i = blockIdx.x * blockDim.x + threadIdx.x;     // 32*256
    float v = h[i] + b1[i & 255];
    h[i] = v > 0.f ? v : 0.f;
}

// ---------------------------------------------------------------------------
// FFN2: h2[m][j] = sum_k h[m][k] * w2[j][k] + b2[j]
// M=32, K=256, Nn=131072.  Streams the 134MB w2 twice (once per 16-row M tile) through
// double-buffered async-LDS staging; fragments via ds_load_b64.
// ---------------------------------------------------------------------------
__global__ void k_ffn2(const float* __restrict__ h, const float* __restrict__ w2,
                       const float* __restrict__ b2, float* __restrict__ h2) {
    __shared__ __align__(16) float sb[2][64 * SB_STRIDE];
    const int lane = threadIdx.x & 31;
    const int n0 = blockIdx.x * 64;
    const int m0 = blockIdx.y * 16;
    const int ml = lane & 15;
    const int kh = (lane >> 4) * 2;
    const float* arow = h + (size_t)(m0 + ml) * 256 + kh;

    auto stage = [&](int buf, int kbeg) {   // 64 w2 rows x K-slab [kbeg,kbeg+64)
#pragma unroll
        for (int rr = 0; rr < 2; ++rr) {
            const int r = lane + 32 * rr;                       // j_rel
            const float* gp = w2 + (size_t)(n0 + r) * 256 + kbeg;
            const unsigned lo = (unsigned)(uintptr_t)(&sb[buf][r * SB_STRIDE]);
#pragma unroll
            for (int c = 0; c < 16; ++c) async_ld16(lo + c * 16u, gp + c * 4);
        }
    };

    stage(0, 0);
    v8f acc0{}, acc1{}, acc2{}, acc3{};
    for (int si = 0; si < 4; ++si) {
        const int kbeg = si * 64;
        if (si < 3) { stage((si + 1) & 1, kbeg + 64); wait_async_le32(); }
        else        { wait_async_0(); }
        const float* sbuf = &sb[si & 1][0];
        for (int k0 = 0; k0 < 64; k0 += 4) {
            v2f a; a.x = arow[kbeg + k0]; a.y = arow[kbeg + k0 + 1];
            const int ko = k0 + kh;
            v2f b;
            b = *(const v2f*)(sbuf + (ml)      * SB_STRIDE + ko); acc0 = wmma4(a, b, acc0);
            b = *(const v2f*)(sbuf + (16 + ml) * SB_STRIDE + ko); acc1 = wmma4(a, b, acc1);
            b = *(const v2f*)(sbuf + (32 + ml) * SB_STRIDE + ko); acc2 = wmma4(a, b, acc2);
            b = *(const v2f*)(sbuf + (48 + ml) * SB_STRIDE + ko); acc3 = wmma4(a, b, acc3);
        }
    }
    const int mrow = m0 + (lane >> 4) * 8;
    const int col  = n0 + ml;
#pragma unroll
    for (int v = 0; v < 8; ++v) {
        float* op = h2 + (size_t)(mrow + v) * 131072 + col;
        op[0]  = acc0[v] + b2[col];
        op[16] = acc1[v] + b2[col + 16];
        op[32] = acc2[v] + b2[col + 32];
        op[48] = acc3[v] + b2[col + 48];
    }
}

// ---------------------------------------------------------------------------
// Output 1x1 conv on yhat = LN2(h2), with BN + residual epilogue:
//   out[nb][o][l] = BN(sum_c w_out[o][c] * yhat[nb][c][l]) + x[nb][o][l]
// M=256 (o), K=128 (c), Nn=1024 (l), batched over nb.  yhat staged via
// double-buffered async-LDS (2 slabs of 64 K-rows).
// ---------------------------------------------------------------------------
__global__ void k_outconv(const float* __restrict__ yhat, const float* __restrict__ w_out,
                          const float* __restrict__ bn_g, const float* __restrict__ bn_b,
                          const float* __restrict__ bn_m, const float* __restrict__ bn_v,
                          const float* __restrict__ x, float* __restrict__ out) {
    __shared__ __align__(16) float sb[2][64 * SB_STRIDE];
    const int lane = threadIdx.x & 31;
    const int n0 = blockIdx.x * 64;      // l tile
    const int m0 = blockIdx.y * 16;      // o tile
    const int nb = blockIdx.z;
    const int ml = lane & 15;
    const int kh = (lane >> 4) * 2;
    const float* B = yhat + (size_t)nb * 131072;       // [128 x 1024]
    const float* arow = w_out + (size_t)(m0 + ml) * 128 + kh;

    auto stage = [&](int buf, int kbeg) {
#pragma unroll
        for (int rr = 0; rr < 2; ++rr) {
            const int r = lane + 32 * rr;                       // k_rel (channel)
            const float* gp = B + (size_t)(kbeg + r) * 1024 + n0;
            const unsigned lo = (unsigned)(uintptr_t)(&sb[buf][r * SB_STRIDE]);
#pragma unroll
            for (int c = 0; c < 16; ++c) async_ld16(lo + c * 16u, gp + c * 4);
        }
    };

    stage(0, 0);
    v8f acc0{}, acc1{}, acc2{}, acc3{};
    for (int si = 0; si < 2; ++si) {
        const int kbeg = si * 64;
        if (si < 1) { stage(1, 64); wait_async_le32(); }
        else        { wait_async_0(); }
        const float* sbuf = &sb[si][0];
        for (int k0 = 0; k0 < 64; k0 += 4) {
            v2f a; a.x = arow[kbeg + k0]; a.y = arow[kbeg + k0 + 1];
            const float* r0 = sbuf + (k0 + kh) * SB_STRIDE + ml;
            const float* r1 = r0 + SB_STRIDE;
            v2f b;
            b.x = r0[0];  b.y = r1[0];  acc0 = wmma4(a, b, acc0);
            b.x = r0[16]; b.y = r1[16]; acc1 = wmma4(a, b, acc1);
            b.x = r0[32]; b.y = r1[32]; acc2 = wmma4(a, b, acc2);
            b.x = r0[48]; b.y = r1[48]; acc3 = wmma4(a, b, acc3);
        }
    }
    const int mrow = m0 + (lane >> 4) * 8;
    const int col  = n0 + ml;
    const float* xn = x + (size_t)nb * 256 * 1024;
    float* on = out + (size_t)nb * 256 * 1024;
#pragma unroll
    for (int v = 0; v < 8; ++v) {
        const int o = mrow + v;
        const float sc = bn_g[o] * rsqrtf(bn_v[o] + EPSF);
        const float mn = bn_m[o], bb = bn_b[o];
        const size_t base = (size_t)o * 1024 + col;
        on[base +  0] = fmaf(acc0[v] - mn, sc, bb) + xn[base +  0];
        on[base + 16] = fmaf(acc1[v] - mn, sc, bb) + xn[base + 16];
        on[base + 32] = fmaf(acc2[v] - mn, sc, bb) + xn[base + 32];
        on[base + 48] = fmaf(acc3[v] - mn, sc, bb) + xn[base + 48];
    }
}

// ---------------------------------------------------------------------------
extern "C" void kernel_launch(void* const* d_in, const int* in_sizes, int n_in,
                              void* d_out, int out_size, void* d_ws, size_t ws_size,
                              hipStream_t stream) {
    const float* x      = (const float*)d_in[0];
    const float* wq     = (const float*)d_in[1];
    const float* wk     = (const float*)d_in[2];
    const float* gamma1 = (const float*)d_in[3];
    const float* beta1  = (const float*)d_in[4];
    const float* w1     = (const float*)d_in[5];
    const float* b1     = (const float*)d_in[6];
    const float* w2     = (const float*)d_in[7];
    const float* b2     = (const float*)d_in[8];
    const float* gamma2 = (const float*)d_in[9];
    const float* beta2  = (const float*)d_in[10];
    const float* w_out  = (const float*)d_in[11];
    const float* bn_g   = (const float*)d_in[12];
    const float* bn_b   = (const float*)d_in[13];
    const float* bn_m   = (const float*)d_in[14];
    const float* bn_v   = (const float*)d_in[15];
    float* out = (float*)d_out;

    // Workspace layout (floats).
    //   region A: q, later h2 (q dead after k_scores)
    //   region B: k, later yhat (k dead after k_scores)
    float* ws = (float*)d_ws;
    float* q     = ws;                       // 32*128*1024 = 4,194,304
    float* h2    = ws;                       //   alias of q
    float* k     = ws + 4194304;             // 4,194,304
    float* yhat  = ws + 4194304;             //   alias of k
    float* s     = ws + 8388608;             // 32*9216    = 294,912
    float* fln   = ws + 8683520;             // 32*9216    = 294,912
    float* h     = ws + 8978432;             // 32*256     = 8,192
    float* mu1   = ws + 8986624;             // 32
    float* rstd1 = ws + 8986656;             // 32
    float* mu2   = ws + 8986688;             // 32
    float* rstd2 = ws + 8986720;             // 32

    // 1) q = wq * x ; k = wk * x   (per-sample [128x256]*[256x1024])
    k_conv1x1<<<dim3(16, 8, 32), 32, 0, stream>>>(wq, x, q);
    k_conv1x1<<<dim3(16, 8, 32), 32, 0, stream>>>(wk, x, k);

    // 2) local attention scores (with 2x fold)
    k_scores<<<dim3(4, 32), 256, 0, stream>>>(q, k, s);

    // 3) LN1: stats + apply -> fln
    k_lnstats<<<32, 256, 0, stream>>>(s, 9216, mu1, rstd1);
    k_ln_affine<<<dim3(36, 32), 256, 0, stream>>>(s, gamma1, beta1, mu1, rstd1, fln, 9216);

    // 4) FFN1 split-K (zero, async-LDS GEMM accumulate, bias+relu)
    k_zero<<<32, 256, 0, stream>>>(h, 32 * 256);
    k_ffn1<<<dim3(4, 2, 36), 32, 0, stream>>>(fln, w1, h);
    k_bias_relu<<<32, 256, 0, stream>>>(h, b1);

    // 5) FFN2 -> h2 (aliases q)
    k_ffn2<<<dim3(2048, 2), 32, 0, stream>>>(h, w2, b2, h2);

    // 6) LN2: stats + apply -> yhat (aliases k)
    k_lnstats<<<32, 256, 0, stream>>>(h2, 131072, mu2, rstd2);
    k_ln_affine<<<dim3(512, 32), 256, 0, stream>>>(h2, gamma2, beta2, mu2, rstd2, yhat, 131072);

    // 7) output conv + BN + residual
    k_outconv<<<dim3(16, 16, 32), 32, 0, stream>>>(yhat, w_out, bn_g, bn_b, bn_m, bn_v, x, out);
}